// Dynamic_33981781246430
// MI455X (gfx1250) — compile-verified
//
#include <hip/hip_runtime.h>
#include <stdint.h>

#define MAXSEG     1024        // max column-segment elements staged per TDM load (f64)
#define DP_BLOCKS  128
#define DP_THREADS 128         // 4 waves/block -> 512 waves total

typedef uint32_t u32x4 __attribute__((ext_vector_type(4)));
typedef uint32_t u32x8 __attribute__((ext_vector_type(8)));

__device__ __forceinline__ uint32_t rfl(uint32_t x) {
    return (uint32_t)__builtin_amdgcn_readfirstlane((int)x);
}

// ---------------------------------------------------------------------------
// TDM: 1-D contiguous copy of `len` f64 elements global -> LDS.
// Descriptor per CDNA5 ISA §8.3 (group0) / §8.4 (group1); groups 2/3 NULL.
// All fields are wave-uniform (forced via readfirstlane) so they land in SGPRs.
// ---------------------------------------------------------------------------
__device__ __forceinline__ void tdm_load_f64_1d(const double* gsrc,
                                                uint32_t lds_byte_addr, int len)
{
    uint64_t ga   = (uint64_t)(uintptr_t)gsrc;
    uint32_t ulen = (uint32_t)len;
    u32x4 g0;
    g0[0] = rfl(1u);                                        // count=1 (valid user D#)
    g0[1] = rfl(lds_byte_addr);                             // lds_addr
    g0[2] = rfl((uint32_t)(ga & 0xFFFFFFFFu));              // global_addr[31:0]
    g0[3] = rfl((uint32_t)((ga >> 32) & 0x01FFFFFFu) | (2u << 30)); // addr[56:32] | type=2
    u32x8 g1;
    g1[0] = rfl(3u << 16);                                  // data_size=3 (8B), mask=0
    g1[1] = rfl((ulen & 0xFFFFu) << 16);                    // tensor_dim0[15:0]
    g1[2] = rfl(((ulen >> 16) & 0xFFFFu) | (1u << 16));     // tensor_dim0[31:16] | tensor_dim1=1
    g1[3] = rfl((ulen & 0xFFFFu) << 16);                    // tensor_dim1[31:16]=0 | tile_dim0=len
    g1[4] = rfl(0u);                                        // tile_dim1=0, tile_dim2=0
    g1[5] = rfl(ulen);                                      // tensor_dim0_stride lo (unused: 1D)
    g1[6] = rfl(0u);
    g1[7] = rfl(0u);
    asm volatile("tensor_load_to_lds %0, %1" :: "s"(g0), "s"(g1) : "memory");
}

// ---------------------------------------------------------------------------
// Kernel 1: per-position prime code p[i] = BASE_PRIMES[argmax_ch feat[0,ch,i,0]]
// ---------------------------------------------------------------------------
__global__ void nuss_prime_kernel(const float* __restrict__ feat,
                                  int* __restrict__ prime, int N)
{
    int i = blockIdx.x * blockDim.x + threadIdx.x;
    if (i >= N) return;
    size_t NN = (size_t)N * N;
    float best = feat[(size_t)i * N];   // ch 0
    int   idx  = 0;
    for (int ch = 1; ch < 4; ++ch) {
        float v = feat[ch * NN + (size_t)i * N];
        if (v > best) { best = v; idx = ch; }   // strict '>' == numpy first-max
    }
    const int P[4] = {2, 3, 5, 7};
    prime[i] = P[idx];
}

// ---------------------------------------------------------------------------
// Kernel 2: build masked symmetric s (f32, exactly matching jnp f32 math),
//           zero dp / dpT (f64), zero output, reset grid barrier.
// ---------------------------------------------------------------------------
__global__ void nuss_init_kernel(const float* __restrict__ con,
                                 const int* __restrict__ prime,
                                 float* __restrict__ s32,
                                 double* __restrict__ dp,
                                 double* __restrict__ dpT,
                                 float* __restrict__ out,
                                 unsigned* __restrict__ bar, int N)
{
    size_t idx = (size_t)blockIdx.x * blockDim.x + threadIdx.x;
    size_t NN = (size_t)N * N;
    if (idx >= NN) return;
    int i = (int)(idx / N), j = (int)(idx % N);
    float c = (con[(size_t)i * N + j] + con[(size_t)j * N + i]) * 0.5f;
    int dij = i - j; if (dij < 0) dij = -dij;
    int prod = prime[i] * prime[j];
    bool ok = (dij >= 4) && (prod == 14 || prod == 15 || prod == 35);
    s32[idx] = ok ? c : 0.0f;
    dp[idx]  = 0.0;
    dpT[idx] = 0.0;
    out[idx] = 0.0f;
    if (idx == 0) *bar = 0u;
}

// ---------------------------------------------------------------------------
// Kernel 3: persistent cooperative Nussinov DP over anti-diagonals.
// One wave per (i,j): column segment dp[k+1][j] comes from dpT via a
// double-buffered TDM load into LDS (TENSORcnt pipelining); row segment
// dp[i][k] is read directly (coalesced, L2-resident). f64 => bit-exact vs ref.
// ---------------------------------------------------------------------------
__global__ void nuss_dp_kernel(const float* __restrict__ s32,
                               double* __restrict__ dp,
                               double* __restrict__ dpT,
                               unsigned* __restrict__ bar, int N)
{
    extern __shared__ double dynlds[];
    const int lane = threadIdx.x & 31;
    const int wave = threadIdx.x >> 5;
    const int wavesPerBlk = blockDim.x >> 5;
    const int gw = blockIdx.x * wavesPerBlk + wave;
    const int GW = gridDim.x * wavesPerBlk;
    double* bufbase = dynlds + (size_t)wave * (2 * MAXSEG);
    const uint32_t lds0 = (uint32_t)(uintptr_t)bufbase;
    const uint32_t lds1 = (uint32_t)(uintptr_t)(bufbase + MAXSEG);

    unsigned target = 0;
    for (int d = 1; d < N; ++d) {
        const int m   = N - d;
        const int seg = d < MAXSEG ? d : MAXSEG;
        int cur = 0;
        int i = gw;
        if (i < m)   // wave-uniform branch
            tdm_load_f64_1d(dpT + (size_t)(i + d) * N + i + 1, lds0, seg);
        for (; i < m; i += GW) {
            const int j = i + d;
            __builtin_amdgcn_s_wait_tensorcnt(0);
            const int inext = i + GW;
            if (inext < m)
                tdm_load_f64_1d(dpT + (size_t)(inext + d) * N + inext + 1,
                                cur ? lds0 : lds1, seg);
            const double* __restrict__ rowp = dp + (size_t)i * N + i;
            const double* __restrict__ buf  = bufbase + (cur ? MAXSEG : 0);
            double best = -1.0e300;
            for (int u = lane; u < d; u += 32) {
                double b = (u < MAXSEG) ? buf[u]
                                        : dpT[(size_t)j * N + (i + 1 + u)];
                best = fmax(best, rowp[u] + b);
            }
            for (int off = 16; off > 0; off >>= 1)
                best = fmax(best, __shfl_xor(best, off));
            // pair term: dp[i+1][j-1] + s[i][j]  (dp zero-init covers d==1)
            double pairv = (double)s32[(size_t)i * N + j]
                         + dp[(size_t)(i + 1) * N + (j - 1)];
            double v = fmax(pairv, best);
            if (lane == 0) {
                dp [(size_t)i * N + j] = v;
                dpT[(size_t)j * N + i] = v;
            }
            cur ^= 1;
        }
        // ---- grid-wide barrier (monotonic counter, agent scope) ----
        target += gridDim.x;
        __threadfence();
        __syncthreads();
        if (threadIdx.x == 0) {
            __hip_atomic_fetch_add(bar, 1u, __ATOMIC_RELEASE, __HIP_MEMORY_SCOPE_AGENT);
            while (__hip_atomic_load(bar, __ATOMIC_ACQUIRE, __HIP_MEMORY_SCOPE_AGENT) < target)
                __builtin_amdgcn_s_sleep(2);
        }
        __syncthreads();
    }
}

// ---------------------------------------------------------------------------
// Kernel 4: traceback, one wave32. All lanes mirror scalar state; argmax over
// k is lane-parallel with (max-value, min-index) shuffle reduction to match
// np.argmax first-max semantics exactly.
// ---------------------------------------------------------------------------
__global__ void nuss_traceback_kernel(const float* __restrict__ s32,
                                      const double* __restrict__ dp,
                                      const double* __restrict__ dpT,
                                      float* __restrict__ out, int N)
{
    __shared__ int st_i[4096];
    __shared__ int st_j[4096];
    const int lane = threadIdx.x;
    const double eps = 1e-9;
    int top = 0;
    st_i[0] = 0; st_j[0] = N - 1; top = 1;   // all lanes write identical value
    while (top > 0) {
        --top;
        int i = st_i[top], j = st_j[top];
        if (j <= i) continue;
        double v = dp[(size_t)i * N + j];
        if (v <= eps) continue;
        if (dp[(size_t)(i + 1) * N + j] >= v - eps) {
            st_i[top] = i + 1; st_j[top] = j; ++top; continue;
        }
        float  sf  = s32[(size_t)i * N + j];
        double sij = (double)sf;
        if (sij > 0.0 && dp[(size_t)(i + 1) * N + (j - 1)] + sij >= v - eps) {
            if (lane == 0) {
                out[(size_t)i * N + j] = sf;
                out[(size_t)j * N + i] = s32[(size_t)j * N + i];
            }
            st_i[top] = i + 1; st_j[top] = j - 1; ++top; continue;
        }
        // k = first argmax over k in [i, j) of dp[i][k] + dp[k+1][j]
        double bv = -1.0e300; int bk = 0x7fffffff;
        for (int k = i + lane; k < j; k += 32) {
            double val = dp[(size_t)i * N + k] + dpT[(size_t)j * N + (k + 1)];
            if (val > bv) { bv = val; bk = k; }
        }
        for (int off = 16; off > 0; off >>= 1) {
            double ov = __shfl_xor(bv, off);
            int    okk = __shfl_xor(bk, off);
            if (ov > bv || (ov == bv && okk < bk)) { bv = ov; bk = okk; }
        }
        int k = bk;
        if (top < 4094) {
            st_i[top] = i;     st_j[top] = k; ++top;
            st_i[top] = k + 1; st_j[top] = j; ++top;
        }
    }
}

// ---------------------------------------------------------------------------
extern "C" void kernel_launch(void* const* d_in, const int* in_sizes, int n_in,
                              void* d_out, int out_size, void* d_ws, size_t ws_size,
                              hipStream_t stream) {
    (void)n_in; (void)out_size; (void)ws_size;
    const int NN = in_sizes[0];          // con: 1*1*N*N
    int N = 1;
    while ((long long)N * N < (long long)NN) ++N;   // N = 1024 for the reference setup

    const float* con  = (const float*)d_in[0];
    const float* feat = (const float*)d_in[1];
    float*       out  = (float*)d_out;

    size_t NNz = (size_t)N * N;
    double*   dp    = (double*)d_ws;                 //  8 MB (N=1024)
    double*   dpT   = dp + NNz;                      //  8 MB
    float*    s32   = (float*)(dpT + NNz);           //  4 MB
    int*      prime = (int*)(s32 + NNz);             //  4 KB
    unsigned* bar   = (unsigned*)(prime + ((N + 63) & ~63));

    nuss_prime_kernel<<<(N + 255) / 256, 256, 0, stream>>>(feat, prime, N);
    nuss_init_kernel<<<(int)((NNz + 255) / 256), 256, 0, stream>>>(
        con, prime, s32, dp, dpT, out, bar, N);
    nuss_dp_kernel<<<DP_BLOCKS, DP_THREADS,
                     (DP_THREADS / 32) * 2 * MAXSEG * sizeof(double), stream>>>(
        s32, dp, dpT, bar, N);
    nuss_traceback_kernel<<<1, 32, 0, stream>>>(s32, dp, dpT, out, N);
}